// TGCN_49426483642626
// MI455X (gfx1250) — compile-verified
//
#include <hip/hip_runtime.h>
#include <hip/hip_bf16.h>

#define N_NODES   50000
#define N_EDGES   1600000
#define ROW_TILES 3125            // 50000 / 16 exactly

typedef float v2f __attribute__((ext_vector_type(2)));
typedef float v8f __attribute__((ext_vector_type(8)));

// ---- workspace layout (float elements) ----
#define DEG_OFF   0               // 50000 floats: deg -> dinv (in place)
#define WC_OFF    50000           // 64*128 fused weights
#define BIAS_OFF  58192           // 128 fused bias
#define U_OFF     58320           // 50000*128  U = x @ Wc
#define S_OFF     6458320         // 50000*128  aggregated messages
// total 12,858,320 floats ~= 51.5 MB

__device__ __forceinline__ float sigm(float t) { return 1.0f / (1.0f + __expf(-t)); }
__device__ __forceinline__ float tanh_fast(float t) {
    float e = __expf(2.0f * t);
    return 1.0f - 2.0f / (e + 1.0f);
}

// 1) zero degree accumulator
__global__ void __launch_bounds__(256) tgcn_zero_deg(float* __restrict__ deg) {
    int i = blockIdx.x * 256 + threadIdx.x;
    if (i < N_NODES) deg[i] = 0.0f;
}

// 2) in-degree via atomics
__global__ void __launch_bounds__(256) tgcn_degree(const int* __restrict__ ei,
                                                   float* __restrict__ deg) {
    int e = blockIdx.x * 256 + threadIdx.x;
    if (e < N_EDGES) {
        int dst = ei[N_EDGES + e];
        __hip_atomic_fetch_add(&deg[dst], 1.0f, __ATOMIC_RELAXED, __HIP_MEMORY_SCOPE_AGENT);
    }
}

// 3) dinv = rsqrt(deg + 2)
__global__ void __launch_bounds__(256) tgcn_dinv(float* __restrict__ deg) {
    int i = blockIdx.x * 256 + threadIdx.x;
    if (i < N_NODES) deg[i] = rsqrtf(deg[i] + 2.0f);
}

// 4) fuse weights: Wc = [W_z @ Wl_z[:64] | W_h @ Wl_h[:64]]  (64 x 128)
//    biasC = [b_z @ Wl_z[:64] + bl_z | b_h @ Wl_h[:64] + bl_h]
__global__ void __launch_bounds__(256) tgcn_prep(
    const float* __restrict__ Wz, const float* __restrict__ bz,
    const float* __restrict__ Wlz, const float* __restrict__ blz,
    const float* __restrict__ Wh, const float* __restrict__ bh,
    const float* __restrict__ Wlh, const float* __restrict__ blh,
    float* __restrict__ Wc, float* __restrict__ biasC)
{
    for (int idx = threadIdx.x; idx < 64 * 128; idx += 256) {
        int k = idx >> 7, j = idx & 127;
        float s = 0.0f;
        if (j < 64) {
            for (int t = 0; t < 64; ++t) s += Wz[k * 64 + t] * Wlz[t * 64 + j];
        } else {
            int jj = j - 64;
            for (int t = 0; t < 64; ++t) s += Wh[k * 64 + t] * Wlh[t * 64 + jj];
        }
        Wc[idx] = s;
    }
    for (int j = threadIdx.x; j < 128; j += 256) {
        float s;
        if (j < 64) {
            s = blz[j];
            for (int t = 0; t < 64; ++t) s += bz[t] * Wlz[t * 64 + j];
        } else {
            int jj = j - 64;
            s = blh[jj];
            for (int t = 0; t < 64; ++t) s += bh[t] * Wlh[t * 64 + jj];
        }
        biasC[j] = s;
    }
}

// 5) U = x @ Wc  via v_wmma_f32_16x16x4_f32; also seed S = 2*dinv^2 * U (self-loop)
//    one wave per 16-row tile, full 128 output columns (8 accumulators)
__global__ void __launch_bounds__(256) tgcn_xw_wmma(
    const float* __restrict__ x,     // [N,64]
    const float* __restrict__ Wc,    // [64,128]
    const float* __restrict__ dinv,  // [N]
    float* __restrict__ U,           // [N,128]
    float* __restrict__ S)           // [N,128]
{
    __shared__ float lw[64 * 128];
    for (int i = threadIdx.x; i < 64 * 128; i += 256) lw[i] = Wc[i];
    __syncthreads();

    const int wave = threadIdx.x >> 5;
    const int lane = threadIdx.x & 31;
    const int tile = blockIdx.x * 8 + wave;
    if (tile >= ROW_TILES) return;

    const int m    = lane & 15;
    const int half = lane >> 4;          // 0: K0/K1, 1: K2/K3
    const int koff = half * 2;
    const int row  = tile * 16 + m;

    v8f acc[8] = {};
    #pragma unroll
    for (int k0 = 0; k0 < 64; k0 += 4) {
        const float2 av = *(const float2*)(x + row * 64 + k0 + koff);
        v2f a; a.x = av.x; a.y = av.y;
        #pragma unroll
        for (int j = 0; j < 8; ++j) {
            const int col = j * 16 + m;
            v2f b;
            b.x = lw[(k0 + koff) * 128 + col];
            b.y = lw[(k0 + koff + 1) * 128 + col];
            acc[j] = __builtin_amdgcn_wmma_f32_16x16x4_f32(
                false, a, false, b, (short)0, acc[j], false, false);
        }
    }

    // D layout: lanes 0-15 hold M=i, lanes 16-31 hold M=8+i; N = lane&15
    const int mbase = tile * 16 + half * 8;
    #pragma unroll
    for (int i = 0; i < 8; ++i) {
        const int node = mbase + i;
        const float dv = dinv[node];
        const float selfw = 2.0f * dv * dv;
        #pragma unroll
        for (int j = 0; j < 8; ++j) {
            const float u = acc[j][i];
            const int idx = node * 128 + j * 16 + m;
            U[idx] = u;
            S[idx] = selfw * u;
        }
    }
}

// 6) edge scatter: S[dst] += dinv[src]*dinv[dst] * U[src]; one wave per edge,
//    each lane covers 4 channels (32 lanes * 4 = 128)
__global__ void __launch_bounds__(256) tgcn_edge_agg(
    const int* __restrict__ ei, const float* __restrict__ dinv,
    const float* __restrict__ U, float* __restrict__ S)
{
    const int wave = blockIdx.x * 8 + (threadIdx.x >> 5);
    if (wave >= N_EDGES) return;
    const int lane = threadIdx.x & 31;
    const int src = ei[wave];
    const int dst = ei[N_EDGES + wave];
    const float w = dinv[src] * dinv[dst];
    const float4 u = *(const float4*)(U + src * 128 + lane * 4);
    float* sp = S + dst * 128 + lane * 4;
    __hip_atomic_fetch_add(sp + 0, w * u.x, __ATOMIC_RELAXED, __HIP_MEMORY_SCOPE_AGENT);
    __hip_atomic_fetch_add(sp + 1, w * u.y, __ATOMIC_RELAXED, __HIP_MEMORY_SCOPE_AGENT);
    __hip_atomic_fetch_add(sp + 2, w * u.z, __ATOMIC_RELAXED, __HIP_MEMORY_SCOPE_AGENT);
    __hip_atomic_fetch_add(sp + 3, w * u.w, __ATOMIC_RELAXED, __HIP_MEMORY_SCOPE_AGENT);
}

// 7) epilogue: H = relu((1-sigmoid(Sz+bz)) * tanh(Sh+bh)); out = H @ W_out + b_out
//    readout GEMM via wmma f32 16x16x4, 45 classes padded to 48 (3 col tiles)
__global__ void __launch_bounds__(256) tgcn_final_wmma(
    const float* __restrict__ S,     // [N,128]
    const float* __restrict__ biasC, // [128]
    const float* __restrict__ Wout,  // [64,45]
    const float* __restrict__ bout,  // [45]
    float* __restrict__ out)         // [N,45]
{
    __shared__ float lw[64 * 48];
    for (int i = threadIdx.x; i < 64 * 48; i += 256) {
        int k = i / 48, c = i % 48;
        lw[i] = (c < 45) ? Wout[k * 45 + c] : 0.0f;
    }
    __syncthreads();

    const int wave = threadIdx.x >> 5;
    const int lane = threadIdx.x & 31;
    const int tile = blockIdx.x * 8 + wave;
    if (tile >= ROW_TILES) return;

    const int m    = lane & 15;
    const int half = lane >> 4;
    const int koff = half * 2;
    const int node = tile * 16 + m;

    // build relu(H) A-fragments: lane covers channels {4k+koff, 4k+koff+1}
    v2f afrag[16];
    #pragma unroll
    for (int kk = 0; kk < 16; ++kk) {
        const int c = kk * 4 + koff;
        const float2 sz = *(const float2*)(S + node * 128 + c);
        const float2 sh = *(const float2*)(S + node * 128 + 64 + c);
        const float2 bz = *(const float2*)(biasC + c);
        const float2 bh = *(const float2*)(biasC + 64 + c);
        float h0 = (1.0f - sigm(sz.x + bz.x)) * tanh_fast(sh.x + bh.x);
        float h1 = (1.0f - sigm(sz.y + bz.y)) * tanh_fast(sh.y + bh.y);
        afrag[kk].x = fmaxf(h0, 0.0f);
        afrag[kk].y = fmaxf(h1, 0.0f);
    }

    v8f acc[3] = {};
    #pragma unroll
    for (int kk = 0; kk < 16; ++kk) {
        #pragma unroll
        for (int j = 0; j < 3; ++j) {
            const int col = j * 16 + m;
            v2f b;
            b.x = lw[(kk * 4 + koff) * 48 + col];
            b.y = lw[(kk * 4 + koff + 1) * 48 + col];
            acc[j] = __builtin_amdgcn_wmma_f32_16x16x4_f32(
                false, afrag[kk], false, b, (short)0, acc[j], false, false);
        }
    }

    const int mbase = tile * 16 + half * 8;
    #pragma unroll
    for (int i = 0; i < 8; ++i) {
        const int nrow = mbase + i;
        #pragma unroll
        for (int j = 0; j < 3; ++j) {
            const int cls = j * 16 + m;
            if (cls < 45) out[nrow * 45 + cls] = acc[j][i] + bout[cls];
        }
    }
}

extern "C" void kernel_launch(void* const* d_in, const int* in_sizes, int n_in,
                              void* d_out, int out_size, void* d_ws, size_t ws_size,
                              hipStream_t stream) {
    const float* x    = (const float*)d_in[0];
    const int*   ei   = (const int*)d_in[1];
    const float* Wz   = (const float*)d_in[2];
    const float* bz   = (const float*)d_in[3];
    const float* Wlz  = (const float*)d_in[4];
    const float* blz  = (const float*)d_in[5];
    // d_in[6..9] = W_r, b_r, Wl_r, bl_r : dead code (H0 == 0 => H*R == 0)
    const float* Wh   = (const float*)d_in[10];
    const float* bh   = (const float*)d_in[11];
    const float* Wlh  = (const float*)d_in[12];
    const float* blh  = (const float*)d_in[13];
    const float* Wout = (const float*)d_in[14];
    const float* bout = (const float*)d_in[15];
    float* out = (float*)d_out;

    float* ws    = (float*)d_ws;
    float* deg   = ws + DEG_OFF;   // becomes dinv in place
    float* Wc    = ws + WC_OFF;
    float* biasC = ws + BIAS_OFF;
    float* U     = ws + U_OFF;
    float* S     = ws + S_OFF;

    tgcn_zero_deg<<<(N_NODES + 255) / 256, 256, 0, stream>>>(deg);
    tgcn_degree<<<(N_EDGES + 255) / 256, 256, 0, stream>>>(ei, deg);
    tgcn_dinv<<<(N_NODES + 255) / 256, 256, 0, stream>>>(deg);
    tgcn_prep<<<1, 256, 0, stream>>>(Wz, bz, Wlz, blz, Wh, bh, Wlh, blh, Wc, biasC);
    tgcn_xw_wmma<<<(ROW_TILES + 7) / 8, 256, 0, stream>>>(x, Wc, deg, U, S);
    tgcn_edge_agg<<<N_EDGES / 8, 256, 0, stream>>>(ei, deg, U, S);
    tgcn_final_wmma<<<(ROW_TILES + 7) / 8, 256, 0, stream>>>(S, biasC, Wout, bout, out);
}